// GraphClassifier_30657476559139
// MI455X (gfx1250) — compile-verified
//
#include <hip/hip_runtime.h>

typedef __attribute__((ext_vector_type(2))) float v2f;
typedef __attribute__((ext_vector_type(8))) float v8f;

#define NNODES 50000
#define NEDGES 800000
#define NGRAPH 128
#define HDIM   128
#define NEG_SLOPE 0.2f

__device__ __forceinline__ float leaky(float e) { return e > 0.f ? e : NEG_SLOPE * e; }

// monotone float <-> uint order transform (for atomicMax on floats)
__device__ __forceinline__ unsigned f2ord(float f) {
    unsigned b = __float_as_uint(f);
    return (b & 0x80000000u) ? ~b : (b | 0x80000000u);
}
__device__ __forceinline__ float ord2f(unsigned b) {
    return (b & 0x80000000u) ? __uint_as_float(b & 0x7fffffffu) : __uint_as_float(~b);
}

__device__ __forceinline__ float wave_sum(float v) {
    #pragma unroll
    for (int off = 16; off > 0; off >>= 1) v += __shfl_xor(v, off, 32);
    return v;
}

// ---------------------------------------------------------------------------
// C[M x 128] = A[M x 128] @ B[128 x 128] using V_WMMA_F32_16X16X4_F32.
// Block = 256 threads = 8 waves. Block b computes rows [16b,16b+16);
// wave w computes the 16x16 tile at columns [16w,16w+16). K-loop: 32 WMMAs.
// ---------------------------------------------------------------------------
__global__ void gemm_nk128_wmma(const float* __restrict__ A,
                                const float* __restrict__ B,
                                float* __restrict__ C) {
    const int wave = threadIdx.x >> 5;
    const int lane = threadIdx.x & 31;
    const int half = lane >> 4;        // 0 or 1
    const int l16  = lane & 15;
    const int row  = blockIdx.x * 16 + l16;   // A row this lane supplies
    const int col  = wave * 16 + l16;         // B/C column this lane supplies

    v8f acc = {0.f, 0.f, 0.f, 0.f, 0.f, 0.f, 0.f, 0.f};
    #pragma unroll
    for (int k = 0; k < HDIM; k += 4) {
        const int ka = k + 2 * half;
        // A fragment: lane holds A[row][ka], A[row][ka+1]  (8B-aligned)
        const float2 a2 = *(const float2*)(A + row * HDIM + ka);
        v2f av; av.x = a2.x; av.y = a2.y;
        // B fragment: lane holds B[ka][col], B[ka+1][col]
        v2f bv; bv.x = B[ka * HDIM + col]; bv.y = B[(ka + 1) * HDIM + col];
        acc = __builtin_amdgcn_wmma_f32_16x16x4_f32(false, av, false, bv,
                                                    (short)0, acc, false, false);
    }
    // C/D layout: VGPR r, lane L -> row = 16*blk + r + 8*(L>>4), col = 16*w + (L&15)
    const int rbase = blockIdx.x * 16 + 8 * half;
    #pragma unroll
    for (int r = 0; r < 8; ++r) C[(rbase + r) * HDIM + col] = acc[r];
}

// ---------------------------------------------------------------------------
__global__ void zero_f(float* p, int n) {
    int i = blockIdx.x * 256 + threadIdx.x;
    if (i < n) p[i] = 0.f;
}

// one wave per node: a_src/a_dst dot products, init max with self-loop logit,
// init deg=1 (self loop), zero GAT accumulator row
__global__ void node_prep(const float* __restrict__ h,
                          const float* __restrict__ att_src,
                          const float* __restrict__ att_dst,
                          float* a_src, float* a_dst, unsigned* m_ord,
                          float* deg, float* acc) {
    const int wave = threadIdx.x >> 5, lane = threadIdx.x & 31;
    const int n = blockIdx.x * 8 + wave;
    if (n >= NNODES) return;
    float s = 0.f, d = 0.f;
    #pragma unroll
    for (int j = 0; j < 4; ++j) {
        int k = lane + 32 * j;
        float v = h[n * HDIM + k];
        s += v * att_src[k];
        d += v * att_dst[k];
        acc[n * HDIM + k] = 0.f;
    }
    s = wave_sum(s); d = wave_sum(d);
    if (lane == 0) {
        a_src[n] = s; a_dst[n] = d;
        m_ord[n] = f2ord(leaky(s + d));   // self-loop logit seeds the max
        deg[n]   = 1.f;                   // self loop seeds degree
    }
}

__global__ void edge_max(const int* __restrict__ src, const int* __restrict__ dst,
                         const float* __restrict__ a_src, const float* __restrict__ a_dst,
                         unsigned* m_ord) {
    int i = blockIdx.x * 256 + threadIdx.x;
    if (i >= NEDGES) return;
    int s = src[i], d = dst[i];
    atomicMax(&m_ord[d], f2ord(leaky(a_src[s] + a_dst[d])));
}

// denom[n] = exp(e_self - m)  (initial value, before edge contributions)
__global__ void node_denom(const float* __restrict__ a_src, const float* __restrict__ a_dst,
                           const unsigned* __restrict__ m_ord, float* denom) {
    int n = blockIdx.x * 256 + threadIdx.x;
    if (n >= NNODES) return;
    denom[n] = __expf(leaky(a_src[n] + a_dst[n]) - ord2f(m_ord[n]));
}

// edge pass: denom[d] += exp(e - m[d]);  deg[d] += 1
__global__ void edge_expsum(const int* __restrict__ src, const int* __restrict__ dst,
                            const float* __restrict__ a_src, const float* __restrict__ a_dst,
                            const unsigned* __restrict__ m_ord, float* denom, float* deg) {
    int i = blockIdx.x * 256 + threadIdx.x;
    if (i >= NEDGES) return;
    int s = src[i], d = dst[i];
    float ex = __expf(leaky(a_src[s] + a_dst[d]) - ord2f(m_ord[d]));
    atomicAdd(&denom[d], ex);
    atomicAdd(&deg[d], 1.f);
}

// one wave per edge: acc[d] += alpha * h[s]
__global__ void gat_scatter(const int* __restrict__ src, const int* __restrict__ dst,
                            const float* __restrict__ a_src, const float* __restrict__ a_dst,
                            const unsigned* __restrict__ m_ord, const float* __restrict__ denom,
                            const float* __restrict__ h, float* acc) {
    int e = blockIdx.x * 8 + (threadIdx.x >> 5);
    if (e >= NEDGES) return;
    int lane = threadIdx.x & 31;
    int s = src[e], d = dst[e];
    float w = __expf(leaky(a_src[s] + a_dst[d]) - ord2f(m_ord[d])) / denom[d];
    #pragma unroll
    for (int j = 0; j < 4; ++j) {
        int k = lane + 32 * j;
        atomicAdd(&acc[d * HDIM + k], w * h[s * HDIM + k]);
    }
}

// acc[n] = relu(acc[n] + alpha_self * h[n] + b1)   (in place: acc becomes h_gat)
__global__ void gat_finalize(const float* __restrict__ h,
                             const float* __restrict__ a_src, const float* __restrict__ a_dst,
                             const unsigned* __restrict__ m_ord, const float* __restrict__ denom,
                             const float* __restrict__ b1, float* acc) {
    const int wave = threadIdx.x >> 5, lane = threadIdx.x & 31;
    const int n = blockIdx.x * 8 + wave;
    if (n >= NNODES) return;
    float w = __expf(leaky(a_src[n] + a_dst[n]) - ord2f(m_ord[n])) / denom[n];
    #pragma unroll
    for (int j = 0; j < 4; ++j) {
        int k = lane + 32 * j;
        acc[n * HDIM + k] = fmaxf(acc[n * HDIM + k] + w * h[n * HDIM + k] + b1[k], 0.f);
    }
}

__global__ void node_dinv(const float* __restrict__ deg, float* dinv) {
    int n = blockIdx.x * 256 + threadIdx.x;
    if (n >= NNODES) return;
    dinv[n] = rsqrtf(deg[n]);   // deg >= 1 due to self loop
}

// one wave per edge: acc[d] += dinv[s]*dinv[d] * h2[s]
__global__ void gcn_scatter(const int* __restrict__ src, const int* __restrict__ dst,
                            const float* __restrict__ dinv, const float* __restrict__ h2,
                            float* acc) {
    int e = blockIdx.x * 8 + (threadIdx.x >> 5);
    if (e >= NEDGES) return;
    int lane = threadIdx.x & 31;
    int s = src[e], d = dst[e];
    float norm = dinv[s] * dinv[d];
    #pragma unroll
    for (int j = 0; j < 4; ++j) {
        int k = lane + 32 * j;
        atomicAdd(&acc[d * HDIM + k], norm * h2[s * HDIM + k]);
    }
}

// relu(acc + dinv^2*h2 + b2), then pool into per-graph sums + counts
__global__ void gcn_final_pool(const float* __restrict__ h2, const float* __restrict__ acc,
                               const float* __restrict__ dinv, const float* __restrict__ b2,
                               const int* __restrict__ batch, float* gsum, float* gcnt) {
    const int wave = threadIdx.x >> 5, lane = threadIdx.x & 31;
    const int n = blockIdx.x * 8 + wave;
    if (n >= NNODES) return;
    float sn = dinv[n] * dinv[n];
    int b = batch[n];
    #pragma unroll
    for (int j = 0; j < 4; ++j) {
        int k = lane + 32 * j;
        float v = fmaxf(acc[n * HDIM + k] + sn * h2[n * HDIM + k] + b2[k], 0.f);
        atomicAdd(&gsum[b * HDIM + k], v);
    }
    if (lane == 0) atomicAdd(&gcnt[b], 1.f);
}

// out[g][c] = (gsum[g]/max(cnt,1)) . Wf[:,c] + bf[c]    (128 graphs x 2 classes)
__global__ void final_fc(const float* __restrict__ gsum, const float* __restrict__ gcnt,
                         const float* __restrict__ Wf, const float* __restrict__ bf,
                         float* __restrict__ out) {
    int t = threadIdx.x;
    int g = t >> 1, c = t & 1;
    float inv = 1.f / fmaxf(gcnt[g], 1.f);
    float s = 0.f;
    for (int k = 0; k < HDIM; ++k) s += (gsum[g * HDIM + k] * inv) * Wf[k * 2 + c];
    out[g * 2 + c] = s + bf[c];
}

// ---------------------------------------------------------------------------
extern "C" void kernel_launch(void* const* d_in, const int* in_sizes, int n_in,
                              void* d_out, int out_size, void* d_ws, size_t ws_size,
                              hipStream_t stream) {
    const float* x       = (const float*)d_in[0];
    const float* W1      = (const float*)d_in[1];
    const float* att_src = (const float*)d_in[2];
    const float* att_dst = (const float*)d_in[3];
    const float* b1      = (const float*)d_in[4];
    const float* W2      = (const float*)d_in[5];
    const float* b2      = (const float*)d_in[6];
    const float* Wf      = (const float*)d_in[7];
    const float* bf      = (const float*)d_in[8];
    const int*   ei      = (const int*)d_in[9];
    const int*   batch   = (const int*)d_in[10];
    const int* src = ei;
    const int* dst = ei + NEDGES;
    float* out = (float*)d_out;

    const size_t NH = (size_t)NNODES * HDIM;
    float*    h     = (float*)d_ws;           // h1, then reused for h2
    float*    acc   = h + NH;                 // GAT acc -> h_gat -> GCN acc
    float*    asrc  = acc + NH;
    float*    adst  = asrc + NNODES;
    unsigned* mord  = (unsigned*)(adst + NNODES);
    float*    denom = (float*)(mord + NNODES);
    float*    deg   = denom + NNODES;
    float*    dinv  = deg + NNODES;
    float*    gsum  = dinv + NNODES;
    float*    gcnt  = gsum + NGRAPH * HDIM;

    // zero pooled sums + counts (contiguous)
    zero_f<<<(NGRAPH * HDIM + NGRAPH + 255) / 256, 256, 0, stream>>>(gsum, NGRAPH * HDIM + NGRAPH);

    // ---- GAT layer ----
    gemm_nk128_wmma<<<NNODES / 16, 256, 0, stream>>>(x, W1, h);                       // h = x@W1
    node_prep<<<NNODES / 8, 256, 0, stream>>>(h, att_src, att_dst, asrc, adst, mord, deg, acc);
    edge_max<<<(NEDGES + 255) / 256, 256, 0, stream>>>(src, dst, asrc, adst, mord);
    node_denom<<<(NNODES + 255) / 256, 256, 0, stream>>>(asrc, adst, mord, denom);
    edge_expsum<<<(NEDGES + 255) / 256, 256, 0, stream>>>(src, dst, asrc, adst, mord, denom, deg);
    gat_scatter<<<NEDGES / 8, 256, 0, stream>>>(src, dst, asrc, adst, mord, denom, h, acc);
    gat_finalize<<<NNODES / 8, 256, 0, stream>>>(h, asrc, adst, mord, denom, b1, acc); // acc = h_gat

    // ---- GCN layer ----
    gemm_nk128_wmma<<<NNODES / 16, 256, 0, stream>>>(acc, W2, h);                     // h = h_gat@W2
    zero_f<<<(int)((NH + 255) / 256), 256, 0, stream>>>(acc, (int)NH);                // acc -> GCN accumulator
    node_dinv<<<(NNODES + 255) / 256, 256, 0, stream>>>(deg, dinv);
    gcn_scatter<<<NEDGES / 8, 256, 0, stream>>>(src, dst, dinv, h, acc);
    gcn_final_pool<<<NNODES / 8, 256, 0, stream>>>(h, acc, dinv, b2, batch, gsum, gcnt);

    // ---- pooled FC ----
    final_fc<<<1, 256, 0, stream>>>(gsum, gcnt, Wf, bf, out);
}